// RGCNLayer_6854767805049
// MI455X (gfx1250) — compile-verified
//
#include <hip/hip_runtime.h>

typedef __attribute__((ext_vector_type(2))) float v2f;
typedef __attribute__((ext_vector_type(8))) float v8f;

#define IN_DIM 128
#define OUT_DIM 128
#define LDS_STRIDE 132 // 128 + 4 pad: bank = (m*132+k)%64 = (4m+k)%64 -> conflict free

// ---------------- binning kernels (relation sort with 16-padding) ----------------

__global__ void rgcn_init_bins(int* __restrict__ counts, int* __restrict__ sorted, int cap) {
    int t = blockIdx.x * blockDim.x + threadIdx.x;
    if (t < 16) counts[t] = 0;
    for (int i = t; i < cap; i += gridDim.x * blockDim.x) sorted[i] = -1;
}

__global__ void rgcn_hist(const int* __restrict__ etype, int E, int* __restrict__ counts) {
    int e = blockIdx.x * blockDim.x + threadIdx.x;
    if (e < E) atomicAdd(&counts[etype[e]], 1);
}

__global__ void rgcn_prefix(const int* __restrict__ counts, int* __restrict__ offsets,
                            int* __restrict__ cursors, int R) {
    if (threadIdx.x == 0 && blockIdx.x == 0) {
        int acc = 0;
        for (int r = 0; r < R; ++r) {
            offsets[r] = acc;
            cursors[r] = acc;
            acc += (counts[r] + 15) & ~15; // pad each bin to multiple of 16
        }
        offsets[R] = acc;
    }
}

__global__ void rgcn_scatter(const int* __restrict__ etype, int E,
                             int* __restrict__ cursors, int* __restrict__ sorted) {
    int e = blockIdx.x * blockDim.x + threadIdx.x;
    if (e < E) {
        int pos = atomicAdd(&cursors[etype[e]], 1);
        sorted[pos] = e;
    }
}

// ---------------- self transform: out = x @ Wself + bias (WMMA f32 16x16x4) ------

__global__ __launch_bounds__(256)
void rgcn_self_gemm(const float* __restrict__ x, const float* __restrict__ W,
                    const float* __restrict__ bias, float* __restrict__ out, int N) {
    __shared__ float xs[16 * LDS_STRIDE];
    const int row0 = blockIdx.x * 16;
    const int tid  = threadIdx.x;
    const bool full = (row0 + 16 <= N); // block-uniform

    // stage 16x128 A tile into LDS
    if (full) {
        for (int i = tid; i < 16 * IN_DIM; i += 256) {
            int m = i >> 7, k = i & 127;
            xs[m * LDS_STRIDE + k] = x[(size_t)(row0 + m) * IN_DIM + k];
        }
    } else {
        for (int i = tid; i < 16 * IN_DIM; i += 256) {
            int m = i >> 7, k = i & 127;
            int row = row0 + m;
            xs[m * LDS_STRIDE + k] = (row < N) ? x[(size_t)row * IN_DIM + k] : 0.0f;
        }
    }
    __syncthreads();

    const int wave = tid >> 5;        // 0..7 -> 16-wide N stripe per wave
    const int lane = tid & 31;
    const int n0   = wave * 16;
    const int mlo  = lane & 15;
    const int koff = (lane >> 4) * 2; // lanes 0-15: K+{0,1}, lanes 16-31: K+{2,3}

    v8f c = {};
#pragma unroll 8
    for (int kk = 0; kk < IN_DIM; kk += 4) {
        const int ka = kk + koff;
        v2f a, b;
        a.x = xs[mlo * LDS_STRIDE + ka];
        a.y = xs[mlo * LDS_STRIDE + ka + 1];
        b.x = W[(size_t)ka * OUT_DIM + n0 + mlo];
        b.y = W[(size_t)(ka + 1) * OUT_DIM + n0 + mlo];
        c = __builtin_amdgcn_wmma_f32_16x16x4_f32(false, a, false, b, (short)0, c, false, false);
    }

    const float bv  = bias[n0 + mlo];
    const int mbase = (lane >> 4) * 8; // C: lanes 0-15 hold M=j, lanes 16-31 hold M=j+8
    float* __restrict__ op = out + (size_t)(row0 + mbase) * OUT_DIM + n0 + mlo;
    if (full) {
#pragma unroll
        for (int j = 0; j < 8; ++j) op[(size_t)j * OUT_DIM] = c[j] + bv;
    } else {
#pragma unroll
        for (int j = 0; j < 8; ++j) {
            if (row0 + mbase + j < N) op[(size_t)j * OUT_DIM] = c[j] + bv;
        }
    }
}

// ---------------- edge messages: out[dst] += x[src] @ W[rel]  (WMMA + atomics) ----

__global__ __launch_bounds__(256)
void rgcn_edge_gemm(const float* __restrict__ x,
                    const int* __restrict__ src, const int* __restrict__ dst,
                    const float* __restrict__ relW,
                    const int* __restrict__ offsets, const int* __restrict__ sorted,
                    float* __restrict__ out, int R) {
    __shared__ float xs[16 * LDS_STRIDE];
    __shared__ int ssrc[16];
    __shared__ int sdst[16];
    __shared__ int srel;

    const int start = blockIdx.x * 16;
    if (start >= offsets[R]) return; // whole block exits together

    const int tid = threadIdx.x;
    if (tid == 0) {
        int r = 0;
        while (r + 1 < R && offsets[r + 1] <= start) ++r;
        srel = r;
    }
    if (tid < 16) {
        int e = sorted[start + tid];
        ssrc[tid] = (e >= 0) ? src[e] : -1;
        sdst[tid] = (e >= 0) ? dst[e] : -1;
    }
    __syncthreads();

    // padding is always the tail of a bin, so a tile is full iff slot 15 is valid
    const bool full = (sdst[15] >= 0); // block-uniform

    // gather 16 source feature rows (padded rows -> zeros)
    if (full) {
        for (int i = tid; i < 16 * IN_DIM; i += 256) {
            int m = i >> 7, k = i & 127;
            xs[m * LDS_STRIDE + k] = x[(size_t)ssrc[m] * IN_DIM + k];
        }
    } else {
        for (int i = tid; i < 16 * IN_DIM; i += 256) {
            int m = i >> 7, k = i & 127;
            int s = ssrc[m];
            xs[m * LDS_STRIDE + k] = (s >= 0) ? x[(size_t)s * IN_DIM + k] : 0.0f;
        }
    }
    __syncthreads();

    const float* __restrict__ W = relW + (size_t)srel * IN_DIM * OUT_DIM; // L2-resident

    const int wave = tid >> 5;
    const int lane = tid & 31;
    const int n0   = wave * 16;
    const int mlo  = lane & 15;
    const int koff = (lane >> 4) * 2;

    v8f c = {};
#pragma unroll 8
    for (int kk = 0; kk < IN_DIM; kk += 4) {
        const int ka = kk + koff;
        v2f a, b;
        a.x = xs[mlo * LDS_STRIDE + ka];
        a.y = xs[mlo * LDS_STRIDE + ka + 1];
        b.x = W[(size_t)ka * OUT_DIM + n0 + mlo];
        b.y = W[(size_t)(ka + 1) * OUT_DIM + n0 + mlo];
        c = __builtin_amdgcn_wmma_f32_16x16x4_f32(false, a, false, b, (short)0, c, false, false);
    }

    const int mbase = (lane >> 4) * 8;
    const int n     = n0 + mlo;
    if (full) {
#pragma unroll
        for (int j = 0; j < 8; ++j) {
            unsafeAtomicAdd(&out[(size_t)sdst[mbase + j] * OUT_DIM + n], c[j]);
        }
    } else {
#pragma unroll
        for (int j = 0; j < 8; ++j) {
            int d = sdst[mbase + j];
            if (d >= 0) unsafeAtomicAdd(&out[(size_t)d * OUT_DIM + n], c[j]);
        }
    }
}

// ---------------- fallback edge kernel (no workspace): plain VALU matvec ---------

__global__ __launch_bounds__(128)
void rgcn_edge_naive(const float* __restrict__ x, const int* __restrict__ src,
                     const int* __restrict__ dst, const int* __restrict__ etype,
                     const float* __restrict__ relW, float* __restrict__ out, int E) {
    __shared__ float xv[IN_DIM];
    int e = blockIdx.x;
    if (e >= E) return;
    int s = src[e], d = dst[e], r = etype[e];
    int tid = threadIdx.x;
    xv[tid] = x[(size_t)s * IN_DIM + tid];
    __syncthreads();
    const float* __restrict__ W = relW + (size_t)r * IN_DIM * OUT_DIM;
    float acc = 0.0f;
#pragma unroll 8
    for (int k = 0; k < IN_DIM; ++k) acc = fmaf(xv[k], W[(size_t)k * OUT_DIM + tid], acc);
    unsafeAtomicAdd(&out[(size_t)d * OUT_DIM + tid], acc);
}

// ---------------- epilogue ----------------

__global__ void rgcn_relu(float* __restrict__ out, int n) {
    int i = blockIdx.x * blockDim.x + threadIdx.x;
    if (i < n) out[i] = fmaxf(out[i], 0.0f);
}

extern "C" void kernel_launch(void* const* d_in, const int* in_sizes, int n_in,
                              void* d_out, int out_size, void* d_ws, size_t ws_size,
                              hipStream_t stream) {
    const float* x     = (const float*)d_in[0];
    const int*   eidx  = (const int*)d_in[1];
    const int*   etype = (const int*)d_in[2];
    const float* relW  = (const float*)d_in[3];
    const float* selfW = (const float*)d_in[4];
    const float* bias  = (const float*)d_in[5];
    float*       out   = (float*)d_out;

    const int N = in_sizes[0] / IN_DIM;
    const int E = in_sizes[2];
    const int R = in_sizes[3] / (IN_DIM * OUT_DIM);
    const int* src = eidx;
    const int* dst = eidx + E;

    // 1) out = x @ Wself + bias
    const int ntiles = (N + 15) / 16;
    rgcn_self_gemm<<<ntiles, 256, 0, stream>>>(x, selfW, bias, out, N);

    // 2) scatter-add relation messages
    const size_t cap  = (size_t)((E + 15) & ~15) + 16u * (size_t)R; // padded bins bound
    const size_t need = (64 + cap) * sizeof(int);
    if (ws_size >= need && R <= 16) {
        int* counts  = (int*)d_ws;     // [0,16)
        int* offsets = counts + 16;    // [16,33)
        int* cursors = counts + 33;    // [33,49)
        int* sorted  = counts + 64;    // [64, 64+cap)
        rgcn_init_bins<<<256, 256, 0, stream>>>(counts, sorted, (int)cap);
        rgcn_hist<<<(E + 255) / 256, 256, 0, stream>>>(etype, E, counts);
        rgcn_prefix<<<1, 32, 0, stream>>>(counts, offsets, cursors, R);
        rgcn_scatter<<<(E + 255) / 256, 256, 0, stream>>>(etype, E, cursors, sorted);
        rgcn_edge_gemm<<<(int)(cap / 16), 256, 0, stream>>>(x, src, dst, relW,
                                                            offsets, sorted, out, R);
    } else {
        rgcn_edge_naive<<<E, 128, 0, stream>>>(x, src, dst, etype, relW, out, E);
    }

    // 3) relu in place
    rgcn_relu<<<(out_size + 255) / 256, 256, 0, stream>>>(out, out_size);
}